// SkeletonLoss_21852793602334
// MI455X (gfx1250) — compile-verified
//
#include <hip/hip_runtime.h>
#include <stdint.h>

#define TPB 256          // 8 wave32s per block
#define EPT 2            // edges per thread
#define EPB (TPB * EPT)  // 512 edges per block

// ---- CDNA5 (gfx1250) async global->LDS staging ---------------------------
// GVS mode: LDS dest from a VGPR byte offset; global addr is 64-bit SGPR base plus 32-bit VGPR offset.
__device__ __forceinline__ void async_load_b128(unsigned lds_byte_off,
                                                const void* sgpr_base,
                                                unsigned vgpr_byte_off) {
  asm volatile("global_load_async_to_lds_b128 %0, %1, %2"
               :
               : "v"(lds_byte_off), "v"(vgpr_byte_off), "s"(sgpr_base)
               : "memory");
}
__device__ __forceinline__ void wait_asynccnt0() {
  asm volatile("s_wait_asynccnt 0" ::: "memory");
}

// partials[blockIdx.x] = sum over this block's edges, all batches, of
// (|p[ia]-p[ib]| - init_len)^2   (unscaled; finalize divides by E)
__global__ __launch_bounds__(TPB) void skel_loss_main(
    const float2* __restrict__ pts,       // [B*N] as float2
    const int4*   __restrict__ sk4,       // [E] int64 pairs viewed as {lo0,hi0,lo1,hi1}
    const float*  __restrict__ init_len,  // [E]
    double*       __restrict__ partials,  // [gridDim.x] in d_ws
    int N, int E, int B) {
  __shared__ int4   sh_edges[EPB];   // 8 KB staged index chunk
  __shared__ double sh_red[TPB];     // 2 KB reduction scratch

  const int tid = threadIdx.x;
  const int blockBase = blockIdx.x * EPB;

  // Stage this block's skeleton chunk into LDS via async DMA (ASYNCcnt path).
  const unsigned shBase = (unsigned)(uintptr_t)(&sh_edges[0]); // LDS byte offset
  #pragma unroll
  for (int k = 0; k < EPT; ++k) {
    const int slot = tid + k * TPB;
    int e = blockBase + slot;
    if (e >= E) e = E - 1;  // clamp: never read OOB; contribution zeroed below
    async_load_b128(shBase + (unsigned)slot * 16u, sk4, (unsigned)e * 16u);
  }
  wait_asynccnt0();
  __syncthreads();

  // Pull this thread's edge indices into registers (reused across all batches).
  int   ia[EPT], ib[EPT];
  float il[EPT], w[EPT];
  #pragma unroll
  for (int k = 0; k < EPT; ++k) {
    const int slot = tid + k * TPB;
    const int e = blockBase + slot;
    const int4 v = sh_edges[slot];
    ia[k] = v.x;                       // low 32 bits of int64 index (N < 2^31)
    ib[k] = v.z;
    const bool valid = (e < E);
    il[k] = valid ? init_len[e] : 0.0f;
    w[k]  = valid ? 1.0f : 0.0f;
  }

  // Batch loop: every gather stays inside one 800 KB L2-resident slab.
  // Unroll x4 -> 16 independent global_load_b64 gathers in flight per wave.
  double acc = 0.0;
  #pragma unroll 4
  for (int b = 0; b < B; ++b) {
    const float2* __restrict__ P = pts + (size_t)b * (size_t)N;
    float2 ps[EPT], pd[EPT];
    #pragma unroll
    for (int k = 0; k < EPT; ++k) { ps[k] = P[ia[k]]; pd[k] = P[ib[k]]; }
    float fsum = 0.0f;
    #pragma unroll
    for (int k = 0; k < EPT; ++k) {
      const float dx  = ps[k].x - pd[k].x;
      const float dy  = ps[k].y - pd[k].y;
      const float len = sqrtf(dx * dx + dy * dy);
      const float t   = len - il[k];
      fsum += w[k] * (t * t);
    }
    acc += (double)fsum;
  }

  // Deterministic block tree-reduction.
  sh_red[tid] = acc;
  __syncthreads();
  for (int s = TPB >> 1; s > 0; s >>= 1) {
    if (tid < s) sh_red[tid] += sh_red[tid + s];
    __syncthreads();
  }
  if (tid == 0) partials[blockIdx.x] = sh_red[0];
}

__global__ __launch_bounds__(TPB) void skel_loss_finalize(
    const double* __restrict__ partials, int nblocks,
    float* __restrict__ out, double invE) {
  __shared__ double sh[TPB];
  double acc = 0.0;
  for (int i = threadIdx.x; i < nblocks; i += TPB) acc += partials[i];
  sh[threadIdx.x] = acc;
  __syncthreads();
  for (int s = TPB >> 1; s > 0; s >>= 1) {
    if (threadIdx.x < s) sh[threadIdx.x] += sh[threadIdx.x + s];
    __syncthreads();
  }
  if (threadIdx.x == 0) out[0] = (float)(sh[0] * invE);
}

extern "C" void kernel_launch(void* const* d_in, const int* in_sizes, int n_in,
                              void* d_out, int out_size, void* d_ws, size_t ws_size,
                              hipStream_t stream) {
  (void)n_in; (void)out_size; (void)ws_size;
  const float2* pts   = (const float2*)d_in[0];   // [B, N, 2] f32
  const int4*   sk4   = (const int4*)d_in[1];     // [E, 2] int64
  const float*  initl = (const float*)d_in[2];    // [E] f32

  const int B = 64;                         // fixed by reference setup
  const int E = in_sizes[2];                // 200000
  const int N = in_sizes[0] / (2 * B);      // 100000
  const int nblocks = (E + EPB - 1) / EPB;  // 391

  double* partials = (double*)d_ws;         // nblocks * 8 bytes of scratch

  skel_loss_main<<<nblocks, TPB, 0, stream>>>(pts, sk4, initl, partials, N, E, B);
  skel_loss_finalize<<<1, TPB, 0, stream>>>(partials, nblocks, (float*)d_out,
                                            1.0 / (double)E);
}